// Gate2_47390669144676
// MI455X (gfx1250) — compile-verified
//
#include <hip/hip_runtime.h>

// ---------------------------------------------------------------------------
// Fused router top-k for MI455X (gfx1250, wave32).
//   router_q = query @ W^T * (1/sqrt(48)), augmented col 48 = 1.0   [4096, 52]
//   router_k = slot_keys @ W^T,            augmented col 48 = mask  [16384, 52]
//   scores   = router_q_aug @ router_k_aug^T   (mask-add folded into K dim,
//              score tensor never materialized: saves 536 MB of HBM traffic)
//   top-8 per query, sorted descending.
//
// Main kernel: one wave per 16-query tile. 16x16 score tiles computed
// *transposed* (M=slot, N=query) with chained V_WMMA_F32_16X16X4_F32 so each
// lane's C-fragment column holds 8 slot scores of ONE query -> exact per-lane
// register top-8 with zero cross-lane traffic in the hot loop. A-fragments are
// software double-buffered (register array) to hide L2 latency; two
// independent accumulator chains break the WMMA RAW chain in half.
// ---------------------------------------------------------------------------

typedef __attribute__((ext_vector_type(2))) float v2f;
typedef __attribute__((ext_vector_type(8))) float v8f;

#define ROUTER_DIM 48
#define STRIDE     52      // 48 + 1 (mask/ones) + 3 zero pad -> 13 K-chunks of 4
#define KCHUNKS    13
#define TOP_K      8
#define D_DIM      256
#define NQ         4096    // B*S = 2*2048
#define NSLOT      16384
#define NEG_INF    (-3.402823466e38f)

// ---------------------------------------------------------------------------
// Projection into augmented layout: out[row][0..47] = scale*dot(X[row],W[r]),
// out[row][48] = aux (mask[row] or 1.0), out[row][49..51] = 0.
// ---------------------------------------------------------------------------
__global__ void proj_kernel(const float* __restrict__ X,
                            const float* __restrict__ W,
                            float* __restrict__ out,
                            int nrows, float scale,
                            const float* __restrict__ aux) {  // null -> 1.0
    int gid = blockIdx.x * blockDim.x + threadIdx.x;
    int row = gid / STRIDE;
    int r   = gid % STRIDE;
    if (row >= nrows) return;
    float val;
    if (r < ROUTER_DIM) {
        const float4* xr = reinterpret_cast<const float4*>(X + row * D_DIM);
        const float4* wr = reinterpret_cast<const float4*>(W + r * D_DIM);
        float acc = 0.0f;
#pragma unroll 8
        for (int d = 0; d < D_DIM / 4; ++d) {
            float4 a = xr[d];
            float4 b = wr[d];
            acc += a.x * b.x + a.y * b.y + a.z * b.z + a.w * b.w;
        }
        val = acc * scale;
    } else if (r == ROUTER_DIM) {
        val = aux ? aux[row] : 1.0f;
    } else {
        val = 0.0f;
    }
    out[row * STRIDE + r] = val;
}

// Sorted-descending top-8 insertion (per lane, registers only).
__device__ __forceinline__ void topk_insert(float (&ts)[TOP_K], int (&ti)[TOP_K],
                                            float s, int idx) {
    if (s <= ts[TOP_K - 1]) return;
    ts[TOP_K - 1] = s;
    ti[TOP_K - 1] = idx;
#pragma unroll
    for (int j = TOP_K - 1; j > 0; --j) {
        if (ts[j] > ts[j - 1]) {
            float fs = ts[j]; ts[j] = ts[j - 1]; ts[j - 1] = fs;
            int   ii = ti[j]; ti[j] = ti[j - 1]; ti[j - 1] = ii;
        }
    }
}

// Load the 13 A-fragments of one slot tile into a register array (clause-able).
__device__ __forceinline__ void load_afrags(v2f (&a)[KCHUNKS],
                                            const float* __restrict__ rk,
                                            int t, int lo, int hi) {
    const float* kb = rk + (t * 16 + lo) * STRIDE + 2 * hi;
#pragma unroll
    for (int c = 0; c < KCHUNKS; ++c)
        a[c] = *reinterpret_cast<const v2f*>(kb + c * 4);
}

// 13 WMMAs in two independent accumulator chains, then top-8 scan.
__device__ __forceinline__ void tile_compute_topk(const v2f (&a)[KCHUNKS],
                                                  const v2f (&b)[KCHUNKS],
                                                  int t, int hi,
                                                  float (&ts)[TOP_K],
                                                  int (&ti)[TOP_K]) {
    v8f acc0 = {};
    v8f acc1 = {};
#pragma unroll
    for (int c = 0; c < KCHUNKS; c += 2)
        acc0 = __builtin_amdgcn_wmma_f32_16x16x4_f32(
            false, a[c], false, b[c], (short)0, acc0, false, false);
#pragma unroll
    for (int c = 1; c < KCHUNKS; c += 2)
        acc1 = __builtin_amdgcn_wmma_f32_16x16x4_f32(
            false, a[c], false, b[c], (short)0, acc1, false, false);

    // Lane holds scores of slots [t*16 + 8*hi, +8) for its query column.
    const int slot0 = t * 16 + 8 * hi;
#pragma unroll
    for (int v = 0; v < 8; ++v) {
        float s = acc0[v] + acc1[v];   // mask already folded in via K-chunk 12
        topk_insert(ts, ti, s, slot0 + v);
    }
}

// ---------------------------------------------------------------------------
// WMMA 16x16x4 f32 operand mapping (wave32):
//   A (16x4, M=slot):  lane L, vgpr v -> A[M=L%16][K=2*(L/16)+v]
//   B (4x16, N=query): lane L, vgpr v -> B[K=2*(L/16)+v][N=L%16]
//   D (16x16):         lane L, vgpr v -> D[M=v+8*(L/16)][N=L%16]
// Both A and B fragments: one 8-byte load at rowbase + c*16B + 8B*(L/16).
// ---------------------------------------------------------------------------
__global__ void __launch_bounds__(128, 1)
score_topk_kernel(const float* __restrict__ rq,     // [NQ][52], pre-scaled, col48=1
                  const float* __restrict__ rk,     // [NSLOT][52], col48=mask
                  int*   __restrict__ out_idx,      // [NQ][8]
                  float* __restrict__ out_score) {  // [NQ][8]
    const int lane  = threadIdx.x & 31;
    const int wave  = threadIdx.x >> 5;
    const int qtile = blockIdx.x * (blockDim.x >> 5) + wave;  // 0..255
    const int lo    = lane & 15;
    const int hi    = lane >> 4;

    // Loop-invariant B fragments: this tile's 16 router_q rows.
    v2f bfrag[KCHUNKS];
    {
        const float* qb = rq + (qtile * 16 + lo) * STRIDE + 2 * hi;
#pragma unroll
        for (int c = 0; c < KCHUNKS; ++c)
            bfrag[c] = *reinterpret_cast<const v2f*>(qb + c * 4);
    }

    float ts[TOP_K];
    int   ti[TOP_K];
#pragma unroll
    for (int j = 0; j < TOP_K; ++j) { ts[j] = NEG_INF; ti[j] = 0; }

    const int ntiles = NSLOT / 16;  // 1024 (even)

    // Software pipeline: double-buffered A-fragment register arrays.
    v2f a0[KCHUNKS];
    v2f a1[KCHUNKS];
    load_afrags(a0, rk, 0, lo, hi);

    for (int t = 0; t < ntiles; t += 2) {
        __builtin_prefetch(rk + (((t + 8) & (ntiles - 1)) * 16 + lo) * STRIDE,
                           0, 1);
        load_afrags(a1, rk, t + 1, lo, hi);              // in flight behind a0 WMMAs
        tile_compute_topk(a0, bfrag, t, hi, ts, ti);
        load_afrags(a0, rk, (t + 2) & (ntiles - 1), lo, hi);  // wrap: harmless dummy
        tile_compute_topk(a1, bfrag, t + 1, hi, ts, ti);
    }

    // Merge with partner lane (lane ^ 16): snapshot first (exactness), then insert.
    float ps[TOP_K];
    int   pi[TOP_K];
#pragma unroll
    for (int j = 0; j < TOP_K; ++j) {
        ps[j] = __shfl(ts[j], lane ^ 16, 32);
        pi[j] = __shfl(ti[j], lane ^ 16, 32);
    }
#pragma unroll
    for (int j = 0; j < TOP_K; ++j) topk_insert(ts, ti, ps[j], pi[j]);

    if (lane < 16) {
        const int row = qtile * 16 + lo;
#pragma unroll
        for (int j = 0; j < TOP_K; ++j) {
            out_idx[row * TOP_K + j]   = ti[j];
            out_score[row * TOP_K + j] = ts[j];
        }
    }
}

// ---------------------------------------------------------------------------
extern "C" void kernel_launch(void* const* d_in, const int* in_sizes, int n_in,
                              void* d_out, int out_size, void* d_ws, size_t ws_size,
                              hipStream_t stream) {
    const float* query     = (const float*)d_in[0];  // [2,2048,256]
    const float* slot_keys = (const float*)d_in[1];  // [16384,256]
    const float* rel_mask  = (const float*)d_in[2];  // [16384]
    const float* W         = (const float*)d_in[3];  // [48,256]

    // Workspace: router_q [4096][52] then router_k [16384][52]  (~4.3 MB).
    float* rq = (float*)d_ws;
    float* rk = rq + (size_t)NQ * STRIDE;

    const float scale = 0.14433756729740643f;  // 1/sqrt(48), folded into router_q

    {
        int total = NQ * STRIDE;
        proj_kernel<<<(total + 255) / 256, 256, 0, stream>>>(
            query, W, rq, NQ, scale, nullptr);           // col48 = 1.0
    }
    {
        int total = NSLOT * STRIDE;
        proj_kernel<<<(total + 255) / 256, 256, 0, stream>>>(
            slot_keys, W, rk, NSLOT, 1.0f, rel_mask);    // col48 = mask
    }

    // Output tuple, flat: top_indices [4096*8] int32, then top_scores [4096*8] f32.
    int*   out_idx   = (int*)d_out;
    float* out_score = (float*)d_out + (size_t)NQ * TOP_K;

    // 256 query tiles of 16 rows -> 256 waves -> 64 blocks x 128 threads (4 waves).
    score_topk_kernel<<<64, 128, 0, stream>>>(rq, rk, out_idx, out_score);
}